// SelfAttention_16767552324199
// MI455X (gfx1250) — compile-verified
//
#include <hip/hip_runtime.h>
#include <stdint.h>

#define BATCH 8
#define CH    64
#define CKQ   8
#define NPIX  4096          // 64*64
#define NB    64            // n-rows staged per block iteration
#define NBLK  (NPIX / NB)   // 64 iterations
#define HSTR  68            // padded LDS row stride in floats (68 % 64 == 4)
#define HBUF  (CH * HSTR)   // floats per LDS buffer

typedef __attribute__((ext_vector_type(2))) float v2f;
typedef __attribute__((ext_vector_type(4))) float v4f;
typedef __attribute__((ext_vector_type(8))) float v8f;

static __device__ __forceinline__ v8f wmma_f32_16x16x4(v2f a, v2f b, v8f c) {
  // D(16x16,f32) += A(16x4,f32) x B(4x16,f32)
  return __builtin_amdgcn_wmma_f32_16x16x4_f32(
      /*neg_a=*/false, a, /*neg_b=*/false, b,
      /*c_mod=*/(short)0, c, /*reuse_a=*/false, /*reuse_b=*/false);
}

static __device__ __forceinline__ float bperm(int byte_addr, float x) {
  return __int_as_float(
      __builtin_amdgcn_ds_bpermute(byte_addr, __float_as_int(x)));
}

// ---------------------------------------------------------------------------
// Kernel 1: fused 1x1-conv projections  f = wq*x, g = wk*x, h = wv*x.
// f and g are stored pair-interleaved: [b][pair(4)][n][2] so the attention
// kernel fetches each K-pair A/B fragment with a single b64 load.
// ---------------------------------------------------------------------------
__global__ __launch_bounds__(256) void sagan_proj_kernel(
    const float* __restrict__ x, const float* __restrict__ wq,
    const float* __restrict__ wk, const float* __restrict__ wv,
    float* __restrict__ f_ws, float* __restrict__ g_ws,
    float* __restrict__ h_ws) {
  __shared__ float wsm[5120];  // wq[512] | wk[512] | wv[4096]
  const int tid = threadIdx.x;
  for (int i = tid; i < 5120; i += 256) {
    float v;
    if (i < 512)        v = wq[i];
    else if (i < 1024)  v = wk[i - 512];
    else                v = wv[i - 1024];
    wsm[i] = v;
  }
  __syncthreads();
  const float* wq_s = wsm;
  const float* wk_s = wsm + 512;
  const float* wv_s = wsm + 1024;

  const int gidx = blockIdx.x * 256 + tid;   // [0, B*N)
  const int b = gidx >> 12;
  const int n = gidx & (NPIX - 1);

  float aq[CKQ], ak[CKQ], av[CH];
#pragma unroll
  for (int k = 0; k < CKQ; ++k) { aq[k] = 0.f; ak[k] = 0.f; }
#pragma unroll
  for (int o = 0; o < CH; ++o) av[o] = 0.f;

  const float* xb = x + (size_t)b * CH * NPIX + n;
  for (int c = 0; c < CH; ++c) {
    const float xv = xb[(size_t)c * NPIX];
#pragma unroll
    for (int k = 0; k < CKQ; ++k) {
      aq[k] = fmaf(wq_s[k * CH + c], xv, aq[k]);
      ak[k] = fmaf(wk_s[k * CH + c], xv, ak[k]);
    }
#pragma unroll
    for (int o = 0; o < CH; ++o) av[o] = fmaf(wv_s[o * CH + c], xv, av[o]);
  }
#pragma unroll
  for (int p = 0; p < 4; ++p) {
    v2f q, g2;
    q.x  = aq[2 * p]; q.y  = aq[2 * p + 1];
    g2.x = ak[2 * p]; g2.y = ak[2 * p + 1];
    *(v2f*)&f_ws[(((size_t)b * 4 + p) * NPIX + n) * 2] = q;
    *(v2f*)&g_ws[(((size_t)b * 4 + p) * NPIX + n) * 2] = g2;
  }
#pragma unroll
  for (int o = 0; o < CH; ++o)
    h_ws[((size_t)b * CH + o) * NPIX + n] = av[o];
}

// ---------------------------------------------------------------------------
// Kernel 2: flash-attention. 8 waves/block share async-staged (double
// buffered) H n-blocks in LDS. Second matmul computes O^T (P as A operand),
// so the P relayout needs only xor-16 half-swaps and the epilogue is b128.
// ---------------------------------------------------------------------------
__global__ __launch_bounds__(256) void sagan_attn_kernel(
    const float* __restrict__ x, const float* __restrict__ gamma,
    const float* __restrict__ f_ws, const float* __restrict__ g_ws,
    const float* __restrict__ h_ws, float* __restrict__ out) {
  __shared__ __align__(16) float h_lds[2][HBUF];

  const int tid  = threadIdx.x;
  const int lane = tid & 31;
  const int wave = tid >> 5;
  const int hf   = lane >> 4;   // lane half
  const int l16  = lane & 15;

  const int b  = blockIdx.x >> 5;                       // 32 blocks per batch
  const int m0 = (((blockIdx.x & 31) << 3) + wave) << 4;

  const float* fb = f_ws + (size_t)b * CKQ * NPIX;  // pair-interleaved
  const float* gb = g_ws + (size_t)b * CKQ * NPIX;  // pair-interleaved
  const float* hb = h_ws + (size_t)b * CH * NPIX;

  // ---- async H staging (cooperative across block, double buffered) ----
  const int crow = tid >> 4;          // base row 0..15 (rounds add +16)
  const int ccol = (tid & 15) << 2;   // float col 0..60 step 4 (16B chunk)
  const uint32_t lds_base = (uint32_t)(uintptr_t)(&h_lds[0][0]);

  auto issue_copy = [&](int nb, int buf) {
    const float* gsrc0 = hb + (size_t)crow * NPIX + nb * NB + ccol;
    const uint32_t ld0 =
        lds_base + 4u * (uint32_t)(buf * HBUF + crow * HSTR + ccol);
#pragma unroll
    for (int rr = 0; rr < 4; ++rr) {   // 4 rounds x 16 rows = 64 rows
      const float* gsrc = gsrc0 + (size_t)(rr * 16) * NPIX;
      const uint32_t ld = ld0 + (uint32_t)(rr * 16 * HSTR * 4);
      asm volatile("global_load_async_to_lds_b128 %0, %1, off"
                   :: "v"(ld), "v"(gsrc) : "memory");
    }
  };

  // ---- G as B-operands (one b64 per K-pair); loop-invariant ----
  v2f gB[2];
#pragma unroll
  for (int t = 0; t < 2; ++t)
    gB[t] = *(const v2f*)&gb[((size_t)(2 * t + hf) * NPIX + m0 + l16) * 2];

  v8f acc[4];  // O^T tile: 16m x 64c (4 c-blocks of 16)
#pragma unroll
  for (int cb = 0; cb < 4; ++cb)
#pragma unroll
    for (int r = 0; r < 8; ++r) acc[cb][r] = 0.f;

  float run_max = -3.402823466e38f;   // per column m = m0 + l16
  float run_sum = 0.f;
  const int sbase = hf << 5;          // bperm base: src lane 8*hf (+r)

  issue_copy(0, 0);
  for (int nb = 0; nb < NBLK; ++nb) {
    const int cur = nb & 1;
    asm volatile("s_wait_asynccnt 0x0" ::: "memory");
    __syncthreads();                       // staged H visible to all waves
    if (nb + 1 < NBLK) issue_copy(nb + 1, cur ^ 1);  // prefetch next block

    const int n0b = nb * NB;

    // ---- scores for 64 rows: 4 sub-tiles x (2 chained K=4 WMMAs) ----
    v8f s[4];
#pragma unroll
    for (int t = 0; t < 4; ++t) {
      const int nn = n0b + t * 16 + l16;
      const v2f fA0 = *(const v2f*)&fb[((size_t)hf * NPIX + nn) * 2];
      const v2f fA1 = *(const v2f*)&fb[((size_t)(2 + hf) * NPIX + nn) * 2];
      v8f z = {};
      z    = wmma_f32_16x16x4(fA0, gB[0], z);   // k = 0..3
      s[t] = wmma_f32_16x16x4(fA1, gB[1], z);   // k = 4..7
    }

    // ---- single online-softmax update per 64 rows (per column m=l16) ----
    float tmax = s[0][0];
#pragma unroll
    for (int t = 0; t < 4; ++t)
#pragma unroll
      for (int r = 0; r < 8; ++r) tmax = fmaxf(tmax, s[t][r]);
    tmax = fmaxf(tmax, __shfl_xor(tmax, 16, 32));
    const float nmax  = fmaxf(run_max, tmax);
    const float scale = __expf(run_max - nmax);

    float tsum = 0.f;
#pragma unroll
    for (int t = 0; t < 4; ++t)
#pragma unroll
      for (int r = 0; r < 8; ++r) {
        s[t][r] = __expf(s[t][r] - nmax);   // reuse s as P
        tsum += s[t][r];
      }
    tsum += __shfl_xor(tsum, 16, 32);
    run_sum = run_sum * scale + tsum;
    run_max = nmax;

    // acc holds O^T: column scale varies per VGPR -> transpose scale vector
    float scale_t[8];
#pragma unroll
    for (int r = 0; r < 8; ++r) scale_t[r] = bperm(sbase + 4 * r, scale);
#pragma unroll
    for (int cb = 0; cb < 4; ++cb)
#pragma unroll
      for (int r = 0; r < 8; ++r) acc[cb][r] *= scale_t[r];

    // ---- O^T += P^T * H^T : per 16-row sub-tile ----
#pragma unroll
    for (int t = 0; t < 4; ++t) {
      // A-operand of P^T: half the rows are in-lane, half in xor-16 partner
      float X[8];
#pragma unroll
      for (int r = 0; r < 8; ++r) X[r] = __shfl_xor(s[t][r], 16, 32);
      v2f pA[4];
#pragma unroll
      for (int v = 0; v < 2; ++v) {
        const float a0 = hf ? X[2 + v]    : s[t][v];      // rows kg*4+2hf+v
        const float a1 = hf ? X[6 + v]    : s[t][4 + v];
        const float a2 = hf ? s[t][2 + v] : X[v];
        const float a3 = hf ? s[t][6 + v] : X[4 + v];
        if (v == 0) { pA[0].x = a0; pA[1].x = a1; pA[2].x = a2; pA[3].x = a3; }
        else        { pA[0].y = a0; pA[1].y = a1; pA[2].y = a2; pA[3].y = a3; }
      }
#pragma unroll
      for (int cb = 0; cb < 4; ++cb) {
        const int c = cb * 16 + l16;
#pragma unroll
        for (int kg = 0; kg < 4; ++kg) {
          const int col = t * 16 + kg * 4 + 2 * hf;
          const v2f hB = *(const v2f*)&h_lds[cur][c * HSTR + col];
          acc[cb] = wmma_f32_16x16x4(pA[kg], hB, acc[cb]);
        }
      }
    }
  }

  // ---- normalize (transposed), gamma-blend with residual, b128 stores ----
  const float inv = 1.0f / run_sum;
  const float gm  = gamma[0];
  float inv_t[8];
#pragma unroll
  for (int r = 0; r < 8; ++r) inv_t[r] = bperm(sbase + 4 * r, inv);

  const size_t base = (size_t)b * CH * NPIX;
#pragma unroll
  for (int cb = 0; cb < 4; ++cb) {
    const int c = cb * 16 + l16;
    const size_t idx = base + (size_t)c * NPIX + m0 + 8 * hf;
    const v4f xlo = *(const v4f*)&x[idx];
    const v4f xhi = *(const v4f*)&x[idx + 4];
    v4f olo, ohi;
#pragma unroll
    for (int r = 0; r < 4; ++r) {
      olo[r] = fmaf(gm, acc[cb][r] * inv_t[r], xlo[r]);
      ohi[r] = fmaf(gm, acc[cb][r + 4] * inv_t[r + 4], xhi[r]);
    }
    *(v4f*)&out[idx]     = olo;
    *(v4f*)&out[idx + 4] = ohi;
  }
}

// ---------------------------------------------------------------------------
extern "C" void kernel_launch(void* const* d_in, const int* in_sizes, int n_in,
                              void* d_out, int out_size, void* d_ws, size_t ws_size,
                              hipStream_t stream) {
  const float* x     = (const float*)d_in[0];
  const float* wq    = (const float*)d_in[1];
  const float* wk    = (const float*)d_in[2];
  const float* wv    = (const float*)d_in[3];
  const float* gamma = (const float*)d_in[4];
  float* out = (float*)d_out;

  float* ws   = (float*)d_ws;
  float* f_ws = ws;                                  // B*8*N  = 1 MB
  float* g_ws = f_ws + (size_t)BATCH * CKQ * NPIX;   // B*8*N  = 1 MB
  float* h_ws = g_ws + (size_t)BATCH * CKQ * NPIX;   // B*64*N = 8 MB

  sagan_proj_kernel<<<BATCH * NPIX / 256, 256, 0, stream>>>(
      x, wq, wk, wv, f_ws, g_ws, h_ws);
  sagan_attn_kernel<<<BATCH * (NPIX / 16) / 8, 256, 0, stream>>>(
      x, gamma, f_ws, g_ws, h_ws, out);
}